// Cross_head_71159018160212
// MI455X (gfx1250) — compile-verified
//
#include <hip/hip_runtime.h>
#include <hip/hip_bf16.h>

typedef __attribute__((ext_vector_type(2))) float v2f;
typedef __attribute__((ext_vector_type(8))) float v8f;
typedef __attribute__((ext_vector_type(4))) unsigned int u32x4;
typedef __attribute__((ext_vector_type(8))) int i32x8;
typedef __attribute__((ext_vector_type(4))) int i32x4;

#define CIN   128
#define COUT  128
#define HWX   5184          // 72*72
#define NVOX  186624        // 36*72*72
#define PP    9
#define NW    576           // HWX / 9
#define LL    2304          // (36/9) * 576
#define PSQ   81

// ---------------------------------------------------------------------------
// Pass 1: invnz[c*LL + l] = 1 / (count_nonzero(att[c,l,0:81]) + 1e-5)
// ---------------------------------------------------------------------------
__global__ __launch_bounds__(256)
void nz_kernel(const float* __restrict__ att, float* __restrict__ invnz) {
    const int wave = (int)((blockIdx.x * blockDim.x + threadIdx.x) >> 5);
    const int lane = (int)(threadIdx.x & 31);
    if (wave >= COUT * LL) return;
    const float* base = att + (long long)wave * PSQ;
    int cnt = 0;
#pragma unroll
    for (int i = 0; i < 3; ++i) {
        const int idx = i * 32 + lane;
        if (idx < PSQ) cnt += (base[idx] != 0.0f) ? 1 : 0;
    }
#pragma unroll
    for (int off = 16; off > 0; off >>= 1)
        cnt += __shfl_xor(cnt, off, 32);
    if (lane == 0) invnz[wave] = 1.0f / ((float)cnt + 1e-5f);
}

// ---------------------------------------------------------------------------
// Pass 2: fused per-voxel channel GEMM (WMMA f32 16x16x4, K=128) + unfold-
// indexed attention mix + LeakyReLU(0.2).
// Workgroup = 8 waves = 128 Cout rows x 64 voxels.
// x tile staged to LDS by the Tensor Data Mover (one descriptor per block,
// hardware row-padding to a 65-dword LDS stride); W fragments in registers.
// ---------------------------------------------------------------------------
__global__ __launch_bounds__(256)
void fused_kernel(const float* __restrict__ x, const float* __restrict__ att,
                  const float* __restrict__ Wm, const float* __restrict__ bias,
                  const float* __restrict__ invnz, float* __restrict__ out) {
    // NOTE: xs must be the ONLY __shared__ object -> static LDS offset 0,
    // which is what the TDM descriptor's lds_addr field assumes.
    __shared__ float xs[CIN * 65];      // 33,280 B (64 data dwords + 1 pad per row)

    const int t      = (int)threadIdx.x;
    const int lane   = t & 31;
    const int waveId = t >> 5;          // 0..7
    const int hi     = lane >> 4;       // 0 | 1 (lane half)
    const int ln     = lane & 15;
    const int vbase  = (int)blockIdx.x * 64;
    const int mbase  = waveId * 16;

    // --- TDM: DMA x tile [128 rows x 64 cols, f32] global -> LDS -----------
    // Row = 64 contiguous floats (256 B); row stride = NVOX floats.
    // pad_enable: after every 64 dwords (pad_interval code 5) insert 1 dword
    // (pad_amount code 0) => LDS row stride 65 dwords (bank-conflict free).
    if (waveId == 0) {
        const unsigned long long ga = (unsigned long long)(const void*)(x + vbase);
        u32x4 g0;
        g0[0] = 1u;                                       // count=1, user desc
        g0[1] = 0u;                                       // lds_addr = 0 (xs)
        g0[2] = (unsigned)(ga & 0xffffffffull);           // global_addr[31:0]
        g0[3] = (unsigned)((ga >> 32) & 0x1ffffffull)     // global_addr[56:32]
              | (2u << 30);                               // type = 2 ("image")
        i32x8 g1;
        g1[0] = (2 << 16)                                 // data_size = 4 B
              | (1 << 20)                                 // pad_enable
              | (5 << 22)                                 // pad_interval: 64 dw
              | (0 << 25);                                // pad_amount: 1 dw
        g1[1] = (int)((NVOX & 0xffff) << 16);             // tensor_dim0[15:0]
        g1[2] = (int)(((unsigned)NVOX >> 16) & 0xffff)    // tensor_dim0[31:16]
              | (CIN << 16);                              // tensor_dim1 = 128
        g1[3] = (64 << 16);                               // tile_dim0 = 64
        g1[4] = CIN;                                      // tile_dim1 = 128 (tile_dim2=0)
        g1[5] = (int)NVOX;                                // tensor_dim0_stride[31:0]
        g1[6] = 0;                                        // stride hi / dim1_stride lo
        g1[7] = 0;
        const i32x4 z4 = {0, 0, 0, 0};
#if defined(__clang_major__) && __clang_major__ >= 23
        const i32x8 z8 = {0, 0, 0, 0, 0, 0, 0, 0};
        __builtin_amdgcn_tensor_load_to_lds(g0, g1, z4, z4, z8, 0);
#else
        __builtin_amdgcn_tensor_load_to_lds(g0, g1, z4, z4, 0);
#endif
        __builtin_amdgcn_s_wait_tensorcnt(0);
    }

    // --- Stage A (W rows mbase..mbase+15) as 32 K-step v2f fragments -------
    // A 16x4 f32 layout: lanes 0-15 -> K={4k,4k+1}; lanes 16-31 -> K={4k+2,4k+3}
    v2f a[32];
    {
        const float* wrow = Wm + (mbase + ln) * CIN + hi * 2;
#pragma unroll
        for (int kk = 0; kk < 32; ++kk) {
            a[kk].x = wrow[kk * 4 + 0];
            a[kk].y = wrow[kk * 4 + 1];
        }
    }
    // Bias pre-broadcast into C layout: VGPR r holds row (mbase + r + 8*hi)
    float bv[8];
#pragma unroll
    for (int r = 0; r < 8; ++r) bv[r] = bias[mbase + r + hi * 8];

    __syncthreads();   // publish TDM-written tile to all waves

    // The TDM builtin carries no pointer operand, so LLVM would otherwise
    // prove the LDS global 'xs' is never stored and fold every read of it to
    // zero (observed in round-2 asm: B operand := v_mov 0). Escaping xs's
    // address into opaque asm with a memory clobber forces the compiler to
    // treat xs as written by the DMA, and fences LDS loads at this point.
    asm volatile("" : : "v"((unsigned long long)(uintptr_t)&xs[0]) : "memory");

    // --- 4 N-subtiles of 16 voxels each ------------------------------------
    for (int nsub = 0; nsub < 4; ++nsub) {
        const int v  = vbase + nsub * 16 + ln;     // voxel for this lane (N = ln)
        const int d  = v / HWX;
        const int hw = v - d * HWX;
        const int l  = (d / PP) * NW + (hw / PP);  // patch index
        const int p  = (d % PP) * PP + (hw % PP);  // kernel position

        v8f acc;
#pragma unroll
        for (int r = 0; r < 8; ++r) acc[r] = bv[r];

        // K=128 in 32 steps of 4, batched: preload 16 B-fragments (clause of
        // ds_load_2addr), then 16 back-to-back WMMAs. B 4x16 layout: N=ln,
        // lane halves hold K pairs {4k,4k+1} / {4k+2,4k+3}.
#pragma unroll
        for (int half = 0; half < 2; ++half) {
            v2f bf[16];
#pragma unroll
            for (int j = 0; j < 16; ++j) {
                const int k = (half * 16 + j) * 4 + hi * 2;
                bf[j].x = xs[(k + 0) * 65 + nsub * 16 + ln];
                bf[j].y = xs[(k + 1) * 65 + nsub * 16 + ln];
            }
#pragma unroll
            for (int j = 0; j < 16; ++j) {
                acc = __builtin_amdgcn_wmma_f32_16x16x4_f32(
                          /*neg_a=*/false, a[half * 16 + j], /*neg_b=*/false, bf[j],
                          /*c_mod=*/(short)0, acc, /*reuse_a=*/false, /*reuse_b=*/false);
            }
        }

        // Epilogue: out = lrelu( y * (1 + y + att/nz), 0.2 )
#pragma unroll
        for (int r = 0; r < 8; ++r) {
            const int   cout = mbase + r + hi * 8;
            const float y    = acc[r];
            const float av   = att[((long long)cout * LL + l) * PSQ + p];
            const float inz  = invnz[cout * LL + l];
            const float o    = y * (1.0f + y + av * inz);
            out[(long long)cout * NVOX + v] = (o >= 0.0f) ? o : 0.2f * o;
        }
    }
}

// ---------------------------------------------------------------------------
extern "C" void kernel_launch(void* const* d_in, const int* in_sizes, int n_in,
                              void* d_out, int out_size, void* d_ws, size_t ws_size,
                              hipStream_t stream) {
    const float* x    = (const float*)d_in[0];   // [1,128,36,72,72]
    const float* att  = (const float*)d_in[1];   // [1,128,2304,81]
    const float* Wm   = (const float*)d_in[2];   // [128,128]
    const float* bias = (const float*)d_in[3];   // [128]
    float*       out  = (float*)d_out;           // [1,128,36,72,72]
    float*       invnz = (float*)d_ws;           // 128*2304 floats = 1.18 MB

    // Pass 1: inverse nonzero counts (one wave per (c,l))
    {
        const int waves  = COUT * LL;             // 294,912
        const int blocks = (waves * 32 + 255) / 256;
        nz_kernel<<<blocks, 256, 0, stream>>>(att, invnz);
    }
    // Pass 2: fused GEMM + attention mix + LeakyReLU
    {
        const int blocks = NVOX / 64;             // 2916
        fused_kernel<<<blocks, 256, 0, stream>>>(x, att, Wm, bias, invnz, out);
    }
}